// MaxIoUAssignerModule_82489141887581
// MI455X (gfx1250) — compile-verified
//
#include <hip/hip_runtime.h>
#include <stdint.h>

// ---------------------------------------------------------------------------
// MaxIoUAssigner for MI455X (gfx1250). Compute-bound (min/max/clip/div IoU
// kernel — no matmul structure, WMMA inapplicable). CDNA5-specific path:
// Tensor Data Mover stages both the GT table (4KB) and each block's prior
// tile (4KB) into LDS asynchronously (TENSORcnt + s_wait_tensorcnt).
//  Pass A: thread-per-GT, loop over LDS prior tile -> 1 global atomicMax/thr.
//  Pass B: thread-per-prior, loop over LDS GT table -> neg/pos/argmax +
//          low-quality override (last GT index wins == reference cand.max).
// IoU >= 0 everywhere, so float ordering/equality == uint-bit ordering and
// both passes share one inlined IoU expression for bitwise-equal recompute.
// ---------------------------------------------------------------------------

typedef unsigned int u32;
typedef __attribute__((ext_vector_type(4))) unsigned int u32x4;
typedef __attribute__((ext_vector_type(4))) int          i32x4;
typedef __attribute__((ext_vector_type(8))) int          i32x8;

#define GTS      256                  // G from the reference
#define GT_DW    (GTS * 4)            // 1024 dwords = 4KB
#define TILE     256                  // priors per block
#define THR      0.5f
#define EPS_IOU  1e-6f

// ---- identical IoU expression used in BOTH passes (bitwise reproducible) --
__device__ __forceinline__ float iou_one(const float4 b, const float4 g,
                                         const float area_b) {
  float area_g = (g.z - g.x) * (g.w - g.y);
  float lx = fmaxf(g.x, b.x);
  float ly = fmaxf(g.y, b.y);
  float rx = fminf(g.z, b.z);
  float ry = fminf(g.w, b.w);
  float w  = fmaxf(rx - lx, 0.0f);
  float h  = fmaxf(ry - ly, 0.0f);
  float inter = w * h;
  float uni   = fmaxf(area_g + area_b - inter, EPS_IOU);
  return inter / uni;
}

// ---- TDM: async 1-row 1D tile (tile_dim0=1024 dwords) into LDS ------------
// tensor_dwords < 1024 on the tail tile: TDM OOB reads return zero.
__device__ __forceinline__ void tdm_load_1d(u32 lds_off, const void* gptr,
                                            u32 tensor_dwords) {
  uint64_t ga = (uint64_t)gptr;
  u32x4 g0;
  g0[0] = 1u;                                    // count=1, user descriptor
  g0[1] = lds_off;                               // lds_addr (byte offset)
  g0[2] = (u32)ga;                               // global_addr[31:0]
  g0[3] = ((u32)(ga >> 32) & 0x01FFFFFFu)        // global_addr[56:32]
          | 0x80000000u;                         // type=2 ("image")
  i32x8 g1;
  g1[0] = 0x00020000;                            // data_size=2 (4 bytes)
  g1[1] = (int)(tensor_dwords << 16);            // tensor_dim0 lo16 @bit48
  g1[2] = (int)((tensor_dwords >> 16)            // tensor_dim0 hi16
                | (1u << 16));                   // tensor_dim1 = 1
  g1[3] = (int)((u32)GT_DW << 16);               // tile_dim0 = 1024
  g1[4] = 1;                                     // tile_dim1=1, tile_dim2=0
  g1[5] = (int)tensor_dwords;                    // tensor_dim0_stride
  g1[6] = 0;
  g1[7] = 0;
  i32x4 gz = {0, 0, 0, 0};
#if __has_builtin(__builtin_amdgcn_tensor_load_to_lds)
  #if defined(__clang_major__) && (__clang_major__ >= 23)
  i32x8 gz8 = {0, 0, 0, 0, 0, 0, 0, 0};
  __builtin_amdgcn_tensor_load_to_lds(g0, g1, gz, gz, gz8, 0);
  #else
  __builtin_amdgcn_tensor_load_to_lds(g0, g1, gz, gz, 0);
  #endif
#else
  (void)g0; (void)g1; (void)gz;                  // cooperative fallback
  for (u32 i = threadIdx.x & 31u; i < tensor_dwords; i += 32u)
    ((__attribute__((address_space(3))) u32*)(uintptr_t)lds_off)[i] =
        ((const u32*)gptr)[i];
#endif
}

__device__ __forceinline__ void wait_tensor0() {
#if __has_builtin(__builtin_amdgcn_s_wait_tensorcnt)
  __builtin_amdgcn_s_wait_tensorcnt(0);
#else
  asm volatile("s_wait_tensorcnt 0x0" ::: "memory");
#endif
}

__device__ __forceinline__ u32 lds_off_of(const void* p) {
  return (u32)(uintptr_t)p;                      // low 32 bits = LDS offset
}

// ---- Pass A: gt_max_bits[t] = max_j iou(gt_t, prior_j), as float bits -----
__global__ void __launch_bounds__(256)
pass_a_gtmax(const float4* __restrict__ bb, const float4* __restrict__ gt,
             u32* __restrict__ gt_max_bits, int N) {
  __shared__ float4 sgt[GTS];                    // GT table, 4KB
  __shared__ float4 spr[TILE];                   // prior tile, 4KB

  int base = (int)blockIdx.x * TILE;
  int cnt  = N - base;
  if (cnt > TILE) cnt = TILE;

  if (threadIdx.x < 32) {                        // one wave issues both DMAs
    tdm_load_1d(lds_off_of(sgt), gt, (u32)GT_DW);
    tdm_load_1d(lds_off_of(spr), bb + base, (u32)cnt * 4u);
    wait_tensor0();
  }
  __syncthreads();

  float4 g    = sgt[threadIdx.x];                // thread t owns gt t
  float  best = 0.0f;                            // IoUs >= 0: identity pad

  for (int p = 0; p < cnt; ++p) {                // uniform LDS broadcast read
    float4 b  = spr[p];
    float  ab = (b.z - b.x) * (b.w - b.y);
    best = fmaxf(best, iou_one(b, g, ab));
  }
  atomicMax(&gt_max_bits[threadIdx.x], __float_as_uint(best));
}

// ---- Pass B: final assignment ---------------------------------------------
__global__ void __launch_bounds__(256)
pass_b_assign(const float4* __restrict__ bb, const float4* __restrict__ gt,
              const u32* __restrict__ gt_max_bits, float* __restrict__ out,
              int N) {
  __shared__ float4 sgt[GTS];
  __shared__ u32    sgm[GTS];

  sgm[threadIdx.x] = gt_max_bits[threadIdx.x];   // one element per thread
  if (threadIdx.x < 32) {
    tdm_load_1d(lds_off_of(sgt), gt, (u32)GT_DW);
    wait_tensor0();
  }
  __syncthreads();

  int j = (int)blockIdx.x * TILE + (int)threadIdx.x;
  if (j >= N) return;

  float4 b  = bb[j];
  float  ab = (b.z - b.x) * (b.w - b.y);

  float best  = -1.0f;                           // >= 0 after loop
  int   bestg = 0;
  int   lq    = -1;

  for (int i = 0; i < GTS; ++i) {
    float v = iou_one(b, sgt[i], ab);            // bitwise == pass A's value
    if (v > best) { best = v; bestg = i; }       // strict '>': first-max argmax
    if (__float_as_uint(v) == sgm[i]) lq = i;    // ties: last i wins
  }

  float a = -1.0f;
  if (best < THR)  a = 0.0f;                     // max_ov >= 0 always holds
  if (best >= THR) a = (float)(bestg + 1);
  if (lq >= 0)     a = (float)(lq + 1);          // low-quality override
  out[j] = a;
}

// ---- ws init (graph replays don't re-poison; must re-zero every call) -----
__global__ void init_gtmax(u32* __restrict__ p) { p[threadIdx.x] = 0u; }

extern "C" void kernel_launch(void* const* d_in, const int* in_sizes, int n_in,
                              void* d_out, int out_size, void* d_ws,
                              size_t ws_size, hipStream_t stream) {
  const float4* bb = (const float4*)d_in[0];     // [N,4] f32 priors
  const float4* gt = (const float4*)d_in[1];     // [256,4] f32 ground truths
  int N = in_sizes[0] / 4;

  u32*   gt_max_bits = (u32*)d_ws;               // 256 * 4B scratch
  float* out         = (float*)d_out;

  init_gtmax<<<1, GTS, 0, stream>>>(gt_max_bits);

  int blocks = (N + TILE - 1) / TILE;
  pass_a_gtmax<<<blocks, 256, 0, stream>>>(bb, gt, gt_max_bits, N);
  pass_b_assign<<<blocks, 256, 0, stream>>>(bb, gt, gt_max_bits, out, N);
}